// SCN1_69810398429356
// MI455X (gfx1250) — compile-verified
//
#include <hip/hip_runtime.h>

#define N_NODES 100000
#define NNZ_E   400000
#define KF      192
#define OUTF    128
#define KF_P    (KF + 4)   // pad 4 floats -> row stride 196 (16B aligned, breaks 64-bank conflict)

typedef __attribute__((ext_vector_type(2))) float v2f;
typedef __attribute__((ext_vector_type(8))) float v8f;

// ---------------------------------------------------------------------------
// Stage 0: zero the SpMM accumulator in workspace (harness poisons d_ws).
// ---------------------------------------------------------------------------
__global__ void scn_zero_kernel(float* __restrict__ p, int n) {
  int i = blockIdx.x * blockDim.x + threadIdx.x;
  int stride = gridDim.x * blockDim.x;
  for (; i < n; i += stride) p[i] = 0.0f;
}

// ---------------------------------------------------------------------------
// Stage 1: COO SpMM  X[row] += vals * x[col]   (one 192-thread block per edge,
// fully coalesced 192-float rows, atomics resolved in L2).
// ---------------------------------------------------------------------------
__global__ __launch_bounds__(KF) void scn_spmm_scatter(
    const int*   __restrict__ row_idx,
    const int*   __restrict__ col_idx,
    const float* __restrict__ vals,
    const float* __restrict__ x,
    float*       __restrict__ X) {
  const int e = blockIdx.x;             // uniform per block -> scalar loads
  const int f = threadIdx.x;
  const int r = row_idx[e];
  const int c = col_idx[e];
  const float v = vals[e];
  atomicAdd(&X[r * KF + f], v * x[c * KF + f]);
}

// ---------------------------------------------------------------------------
// Stage 2: dense projection  out = X @ theta  via V_WMMA_F32_16X16X4_F32.
// Block = 256 threads = 8 waves; block owns a 16-row strip of X (staged in
// LDS), wave w owns output cols [16w, 16w+16). 48 chained WMMAs over K=192.
// ---------------------------------------------------------------------------
__global__ __launch_bounds__(256) void scn_gemm_wmma(
    const float* __restrict__ X,
    const float* __restrict__ theta,
    float*       __restrict__ out) {
  __shared__ float sX[16 * KF_P];       // 16 rows x 196 floats = 12.25 KB

  const int tid = threadIdx.x;
  const int m0  = blockIdx.x * 16;      // row-strip base (N = 6250 * 16 exact)

  // Cooperative float4 load of the 16x192 strip into LDS (768 float4 total).
  {
    const float4* src = (const float4*)(X + (size_t)m0 * KF);
    float4* dst = (float4*)sX;
    #pragma unroll
    for (int i = 0; i < 3; ++i) {
      int idx = tid + i * 256;          // 0 .. 767
      int r = idx / (KF / 4);           // row 0..15
      int c = idx % (KF / 4);           // float4 col 0..47
      dst[r * (KF_P / 4) + c] = src[r * (KF / 4) + c];
    }
  }
  __syncthreads();

  const int wave = tid >> 5;            // 0..7 -> output col tile
  const int lane = tid & 31;
  const int m    = lane & 15;           // row within tile / col within tile
  const int g    = lane >> 4;           // K-half select
  const int n0   = wave * 16;

  v8f acc = {};
  #pragma unroll 4
  for (int k0 = 0; k0 < KF; k0 += 4) {
    v2f a, b;
    const int ka = k0 + 2 * g;
    // A fragment: A[m][ka+j] from LDS
    a.x = sX[m * KF_P + ka];
    a.y = sX[m * KF_P + ka + 1];
    // B fragment: theta[ka+j][n0+m] from global (tiny matrix, L2/WGP$ resident)
    b.x = theta[(ka    ) * OUTF + n0 + m];
    b.y = theta[(ka + 1) * OUTF + n0 + m];
    acc = __builtin_amdgcn_wmma_f32_16x16x4_f32(
        /*neg_a=*/false, a, /*neg_b=*/false, b,
        /*c_mod=*/(short)0, acc, /*reuse_a=*/false, /*reuse_b=*/false);
  }

  // D layout: VGPR r, lane (m,g) -> out row m0 + r + 8*g, col n0 + m
  #pragma unroll
  for (int r = 0; r < 8; ++r) {
    out[(size_t)(m0 + r + 8 * g) * OUTF + n0 + m] = acc[r];
  }
}

// ---------------------------------------------------------------------------
extern "C" void kernel_launch(void* const* d_in, const int* in_sizes, int n_in,
                              void* d_out, int out_size, void* d_ws, size_t ws_size,
                              hipStream_t stream) {
  const int*   row_idx = (const int*)d_in[0];
  const int*   col_idx = (const int*)d_in[1];
  const float* vals    = (const float*)d_in[2];
  const float* x       = (const float*)d_in[3];
  const float* theta   = (const float*)d_in[4];
  float* outp = (float*)d_out;
  float* Xws  = (float*)d_ws;           // N_NODES * KF floats = 76.8 MB scratch

  const int total = N_NODES * KF;
  scn_zero_kernel<<<2048, 256, 0, stream>>>(Xws, total);
  scn_spmm_scatter<<<NNZ_E, KF, 0, stream>>>(row_idx, col_idx, vals, x, Xws);
  scn_gemm_wmma<<<N_NODES / 16, 256, 0, stream>>>(Xws, theta, outp);
}